// LindeBuzoGrayAlgorithm_60997125538227
// MI455X (gfx1250) — compile-verified
//
#include <hip/hip_runtime.h>
#include <stdint.h>

// ---------------------------------------------------------------------------
// LBG vector quantization for MI455X (gfx1250, wave32, WMMA)
// x: 131072 x 64 f32, K=256, 8 splits x 5 Lloyd iterations.
// E-step GEMM uses V_WMMA_F32_16X16X4_F32 (fp32 required: perturbations ~1e-5).
// Two point-tiles share each B fetch -> two independent WMMA chains.
// ---------------------------------------------------------------------------

typedef float v8f __attribute__((ext_vector_type(8)));
typedef float v2f __attribute__((ext_vector_type(2)));

#define LBG_N 131072
#define LBG_D 64
#define LBG_K 256
#define PERTURB 1e-5f
#define EPSC 1e-5f

// ---------------- Threefry-2x32 (JAX PRNG), host + device ------------------
#define TF_ROUND(r) { x0 += x1; x1 = (x1 << (r)) | (x1 >> (32 - (r))); x1 ^= x0; }
__host__ __device__ static inline void tf2x32(uint32_t k0, uint32_t k1,
                                              uint32_t c0, uint32_t c1,
                                              uint32_t& o0, uint32_t& o1) {
  uint32_t ks2 = 0x1BD11BDAu ^ k0 ^ k1;
  uint32_t x0 = c0 + k0;
  uint32_t x1 = c1 + k1;
  TF_ROUND(13) TF_ROUND(15) TF_ROUND(26) TF_ROUND(6)
  x0 += k1;  x1 += ks2 + 1u;
  TF_ROUND(17) TF_ROUND(29) TF_ROUND(16) TF_ROUND(24)
  x0 += ks2; x1 += k0 + 2u;
  TF_ROUND(13) TF_ROUND(15) TF_ROUND(26) TF_ROUND(6)
  x0 += k0;  x1 += k1 + 3u;
  TF_ROUND(17) TF_ROUND(29) TF_ROUND(16) TF_ROUND(24)
  x0 += k1;  x1 += ks2 + 4u;
  TF_ROUND(13) TF_ROUND(15) TF_ROUND(26) TF_ROUND(6)
  x0 += ks2; x1 += k0 + 5u;
  o0 = x0; o1 = x1;
}
#undef TF_ROUND

__device__ static inline float erfinv_f(float u) {
  float w = -logf((1.0f - u) * (1.0f + u));
  float p;
  if (w < 5.0f) {
    w -= 2.5f;
    p = 2.81022636e-08f;
    p = 3.43273939e-07f + p * w;  p = -3.5233877e-06f  + p * w;
    p = -4.39150654e-06f + p * w; p = 0.00021858087f   + p * w;
    p = -0.00125372503f  + p * w; p = -0.00417768164f  + p * w;
    p = 0.246640727f     + p * w; p = 1.50140941f      + p * w;
  } else {
    w = sqrtf(w) - 3.0f;
    p = -0.000200214257f;
    p = 0.000100950558f + p * w;  p = 0.00134934322f   + p * w;
    p = -0.00367342844f + p * w;  p = 0.00573950773f   + p * w;
    p = -0.0076224613f  + p * w;  p = 0.00943887047f   + p * w;
    p = 1.00167406f     + p * w;  p = 2.83297682f      + p * w;
  }
  return p * u;
}

// JAX random.normal element j of a flat draw of `total` elements (total even):
// threefry counter halves pairing, bits -> [1,2) -> uniform(lo,1) -> sqrt2*erfinv
__device__ static inline float tf_normal(uint32_t k0, uint32_t k1,
                                         uint32_t j, uint32_t total) {
  uint32_t h = total >> 1;
  uint32_t o0, o1, bits;
  if (j < h) { tf2x32(k0, k1, j, j + h, o0, o1); bits = o0; }
  else       { tf2x32(k0, k1, j - h, j, o0, o1); bits = o1; }
  float f = __uint_as_float((bits >> 9) | 0x3f800000u);   // [1,2)
  const float lo = -0.99999994f, hi = 1.0f;
  float u = f * (hi - lo) + (lo - (hi - lo));             // [lo, hi)
  u = fmaxf(lo, u);
  return 1.41421356237f * erfinv_f(u);
}

// ---------------------------- utility kernels ------------------------------
__global__ void zero_f(float* __restrict__ p, int n) {
  int j = blockIdx.x * 256 + threadIdx.x;
  if (j < n) p[j] = 0.0f;
}

// per-dim sums (for mean) + total sum of squares, LDS-staged
__global__ void lbg_meansq(const float* __restrict__ x,
                           float* __restrict__ macc, float* __restrict__ sxx) {
  __shared__ float sm[64];
  __shared__ float sr[256];
  int tid = threadIdx.x;
  if (tid < 64) sm[tid] = 0.0f;
  __syncthreads();
  float sq = 0.0f;
  for (size_t j = (size_t)blockIdx.x * 256 + tid; j < (size_t)LBG_N * LBG_D; j += 65536) {
    float v = x[j];
    atomicAdd(&sm[(int)(j & 63)], v);
    sq += v * v;
  }
  sr[tid] = sq;
  __syncthreads();
  for (int off = 128; off >= 1; off >>= 1) {
    if (tid < off) sr[tid] += sr[tid + off];
    __syncthreads();
  }
  if (tid < 64) atomicAdd(&macc[tid], sm[tid]);
  if (tid == 0) atomicAdd(sxx, sr[0]);
}

// cb[0] = mean, cb[1..255] = 1e10 (reference init); state = {prev=inf, dist=inf, done=0}
__global__ void lbg_initcb(float* __restrict__ cb, const float* __restrict__ macc,
                           float* __restrict__ state) {
  int j = blockIdx.x * 256 + threadIdx.x;
  if (j < LBG_K * LBG_D)
    cb[j] = (j < LBG_D) ? (macc[j] / (float)LBG_N) : 1.0e10f;
  if (j == 0) {
    state[0] = __builtin_huge_valf();  // prev distance
    state[1] = __builtin_huge_valf();  // distance
    state[2] = 0.0f;                   // done
  }
}

// codebook split: cb[curr+i] = cb[i] - r ; cb[i] += r ; reset loop state
__global__ void lbg_split(float* __restrict__ cb, float* __restrict__ state,
                          int curr_old, unsigned kr0, unsigned kr1) {
  int total = curr_old * LBG_D;
  for (int j = threadIdx.x; j < total; j += 256) {
    float r = PERTURB * tf_normal(kr0, kr1, (unsigned)j, (unsigned)total);
    float v = cb[j];
    cb[total + j] = v - r;   // rows are contiguous: row (curr_old + i) dim d
    cb[j] = v + r;
  }
  if (threadIdx.x == 0) {
    state[0] = state[1];     // prev = distance
    state[2] = 0.0f;         // done = false
  }
}

// per-centroid squared norms (padded 1e10 rows -> ~6.4e21, never win argmin)
__global__ void lbg_cnorm(const float* __restrict__ cb, float* __restrict__ cnorm) {
  int k = threadIdx.x;
  float s = 0.0f;
  #pragma unroll 8
  for (int d = 0; d < LBG_D; ++d) { float v = cb[k * LBG_D + d]; s += v * v; }
  cnorm[k] = s;
}

// ---------------------------------------------------------------------------
// Fused E-step: two 16-point tiles share each B fetch (two independent WMMA
// accumulator chains -> fills WMMA RAW bubbles, halves B traffic), cross-lane
// argmin via wave32 shuffles, LDS-staged segment sums/counts, distortion
// reduction. Block: 256 threads (8 waves) x 2 tile-pairs = 512 pts/block.
// Grid: 256 blocks (exactly N). EXEC is all-ones everywhere around WMMA.
// ---------------------------------------------------------------------------
__global__ void lbg_estep(const float* __restrict__ x, const float* __restrict__ cb,
                          const float* __restrict__ cnorm, float* __restrict__ g_sums,
                          float* __restrict__ g_counts, float* __restrict__ g_dist,
                          int curr, int curr16) {
  extern __shared__ float smem[];
  float* s_sums   = smem;                          // curr16*64
  float* s_counts = s_sums + curr16 * 64;          // curr16
  float* s_val    = s_counts + curr16;             // 8 waves * 16
  int*   s_idx    = (int*)(s_val + 128);           // 8 waves * 16
  float* s_red    = (float*)(s_idx + 128);         // 256

  const int tid  = threadIdx.x;
  const int wave = tid >> 5;
  const int lane = tid & 31;
  const int half = lane >> 4;      // 16x16 C/D layout: lanes 16-31 hold rows M+8
  const int lm   = lane & 15;
  const int ntiles = curr16 >> 4;

  for (int j = tid; j < curr16 * 64; j += 256) s_sums[j] = 0.0f;
  for (int j = tid; j < curr16; j += 256) s_counts[j] = 0.0f;
  __syncthreads();

  float dist_acc = 0.0f;
  const int block_base = blockIdx.x * 512;

  for (int pp = 0; pp < 2; ++pp) {                 // 2 pairs of point-tiles
    const int row0 = block_base + (wave * 4 + pp * 2) * 16;
    const float* xr0 = x + (size_t)(row0 + lm) * LBG_D + 2 * half;
    const float* xr1 = xr0 + 16 * LBG_D;
    if (pp == 0) {                                  // global_prefetch_b8
      __builtin_prefetch(xr0 + 32 * LBG_D, 0, 0);
      __builtin_prefetch(xr1 + 32 * LBG_D, 0, 0);
    }

    // A fragments: 16x4 f32 per K-step; lane lm = row, half selects K pair
    v2f a0[16], a1[16];
    #pragma unroll
    for (int s = 0; s < 16; ++s) {
      a0[s] = *(const v2f*)(xr0 + 4 * s);
      a1[s] = *(const v2f*)(xr1 + 4 * s);
    }

    float minv0[8], minv1[8];
    int   mini0[8], mini1[8];
    #pragma unroll
    for (int r = 0; r < 8; ++r) {
      minv0[r] = 3.4e38f; mini0[r] = 0;
      minv1[r] = 3.4e38f; mini1[r] = 0;
    }

    for (int ct = 0; ct < ntiles; ++ct) {
      const int col = ct * 16 + lm;
      const float* cr = cb + (size_t)col * LBG_D + 2 * half;
      v8f acc0 = {0.f, 0.f, 0.f, 0.f, 0.f, 0.f, 0.f, 0.f};
      v8f acc1 = {0.f, 0.f, 0.f, 0.f, 0.f, 0.f, 0.f, 0.f};
      #pragma unroll
      for (int s = 0; s < 16; ++s) {
        v2f b = *(const v2f*)(cr + 4 * s);
        acc0 = __builtin_amdgcn_wmma_f32_16x16x4_f32(
            false, a0[s], false, b, (short)0, acc0, false, false);
        acc1 = __builtin_amdgcn_wmma_f32_16x16x4_f32(
            false, a1[s], false, b, (short)0, acc1, false, false);
      }
      const float cn = cnorm[col];
      #pragma unroll
      for (int r = 0; r < 8; ++r) {
        float sv0 = fmaf(-2.0f, acc0[r], cn);  // ||c||^2 - 2 x.c (argmin-equiv)
        float sv1 = fmaf(-2.0f, acc1[r], cn);
        if (sv0 < minv0[r]) { minv0[r] = sv0; mini0[r] = col; }
        if (sv1 < minv1[r]) { minv1[r] = sv1; mini1[r] = col; }
      }
    }

    // argmin across 16-lane halves (rows r / r+8), tie -> lowest index (jnp.argmin)
    #pragma unroll
    for (int r = 0; r < 8; ++r) {
      #pragma unroll
      for (int msk = 8; msk >= 1; msk >>= 1) {
        float ov0 = __shfl_xor(minv0[r], msk, 32);
        int   oi0 = __shfl_xor(mini0[r], msk, 32);
        if (ov0 < minv0[r] || (ov0 == minv0[r] && oi0 < mini0[r])) {
          minv0[r] = ov0; mini0[r] = oi0;
        }
        float ov1 = __shfl_xor(minv1[r], msk, 32);
        int   oi1 = __shfl_xor(mini1[r], msk, 32);
        if (ov1 < minv1[r] || (ov1 == minv1[r] && oi1 < mini1[r])) {
          minv1[r] = ov1; mini1[r] = oi1;
        }
      }
    }

    // scatter both tiles: exchange per-row (val,idx) through LDS, then each
    // (lm, lm+16) lane pair adds its 2x16 held dims of row lm to cluster k.
    #pragma unroll
    for (int t = 0; t < 2; ++t) {
      if (lm == 0) {
        #pragma unroll
        for (int r = 0; r < 8; ++r) {
          s_val[wave * 16 + half * 8 + r] = t ? minv1[r] : minv0[r];
          s_idx[wave * 16 + half * 8 + r] = t ? mini1[r] : mini0[r];
        }
      }
      __builtin_amdgcn_wave_barrier();
      asm volatile("s_wait_dscnt 0" ::: "memory");

      const int k = s_idx[wave * 16 + lm];
      if (half == 0) {
        dist_acc += s_val[wave * 16 + lm];
        atomicAdd(&s_counts[k], 1.0f);          // ds_add_f32
      }
      float* dstrow = s_sums + k * 64 + 2 * half;
      #pragma unroll
      for (int s = 0; s < 16; ++s) {
        atomicAdd(dstrow + 4 * s,     t ? a1[s].x : a0[s].x);
        atomicAdd(dstrow + 4 * s + 1, t ? a1[s].y : a0[s].y);
      }
      __builtin_amdgcn_wave_barrier();
      asm volatile("s_wait_dscnt 0" ::: "memory");
    }
  }

  // block-reduce distortion, single global atomic per block
  s_red[tid] = dist_acc;
  __syncthreads();
  for (int off = 128; off >= 1; off >>= 1) {
    if (tid < off) s_red[tid] += s_red[tid + off];
    __syncthreads();
  }
  if (tid == 0) atomicAdd(g_dist, s_red[0]);

  // flush LDS segment sums/counts (only live rows k < curr)
  for (int j = tid; j < curr * 64; j += 256) atomicAdd(&g_sums[j], s_sums[j]);
  for (int j = tid; j < curr; j += 256) atomicAdd(&g_counts[j], s_counts[j]);
}

// ---------------------------------------------------------------------------
// M-step + convergence bookkeeping (single block; mirrors the jax scan body)
// ---------------------------------------------------------------------------
__global__ void lbg_mstep(float* __restrict__ cb, const float* __restrict__ sums,
                          const float* __restrict__ counts, const float* __restrict__ dacc,
                          float* __restrict__ state, const float* __restrict__ sxx,
                          int curr, int n, unsigned sk0, unsigned sk1) {
  __shared__ float s_cnt[256];
  __shared__ int   s_arg[256];
  __shared__ float s_centm[64];
  __shared__ float s_corr[64];
  __shared__ float s_nempty;
  __shared__ int   s_stop;
  const int tid = threadIdx.x;

  if (tid == 0) {
    s_nempty = 0.0f;
    float prev = state[0], dist = state[1], done = state[2];
    float d_new = (sxx[0] + dacc[0]) / (float)LBG_N;
    float change = fabsf(prev - d_new);
    int conv = (n > 0) && (change / (d_new + 1e-16f) < EPSC);
    int stop = (done != 0.0f) || conv;
    s_stop = stop;
    state[1] = (done != 0.0f) ? dist : d_new;   // dist_out uses old done
    state[0] = stop ? prev : d_new;             // prev_out
    state[2] = stop ? 1.0f : 0.0f;              // done_out
  }
  float c = (tid < curr) ? counts[tid] : -1.0f;
  s_cnt[tid] = c;
  s_arg[tid] = tid;
  __syncthreads();
  if (tid < curr && c < 1.0f) atomicAdd(&s_nempty, 1.0f);
  // argmax(counts), first-occurrence tie-break like jnp.argmax
  for (int off = 128; off >= 1; off >>= 1) {
    if (tid < off) {
      float b = s_cnt[tid + off];
      if (b > s_cnt[tid] || (b == s_cnt[tid] && s_arg[tid + off] < s_arg[tid])) {
        s_cnt[tid] = b; s_arg[tid] = s_arg[tid + off];
      }
    }
    __syncthreads();
  }
  const int m = s_arg[0];
  const unsigned total = (unsigned)curr * 64u;
  if (tid < 64) {
    s_centm[tid] = sums[m * 64 + tid] / fmaxf(counts[m], 1.0f);
    float acc = 0.0f;
    for (int k = 0; k < curr; ++k)
      if (counts[k] < 1.0f)
        acc += PERTURB * tf_normal(sk0, sk1, (unsigned)(k * 64 + tid), total);
    s_corr[tid] = acc;   // raw sum of r over empty rows
  }
  __syncthreads();
  const int cntE = (int)(s_nempty + 0.5f);
  if (!s_stop) {
    for (int j = tid; j < curr * 64; j += 256) {
      int k = j >> 6, d = j & 63;
      float ck = counts[k];
      float out;
      if (ck >= 1.0f) out = sums[j] / ck;  // MIN_DATA=1 -> max(n,1)=n here
      else            out = s_centm[d] - PERTURB * tf_normal(sk0, sk1, (unsigned)j, total);
      if (k == m && cntE > 0) out += s_corr[d] / fmaxf((float)cntE, 1.0f);
      cb[j] = out;
    }
  }
}

__global__ void lbg_finalize(const float* __restrict__ cb,
                             const float* __restrict__ state,
                             float* __restrict__ out) {
  int j = blockIdx.x * 256 + threadIdx.x;
  if (j < LBG_K * LBG_D) out[j] = cb[j];
  if (j == LBG_K * LBG_D) out[j] = state[1];
}

// ---------------------------------------------------------------------------
extern "C" void kernel_launch(void* const* d_in, const int* in_sizes, int n_in,
                              void* d_out, int out_size, void* d_ws, size_t ws_size,
                              hipStream_t stream) {
  (void)in_sizes; (void)n_in; (void)out_size; (void)ws_size;
  const float* x = (const float*)d_in[0];
  float* ws     = (float*)d_ws;
  float* cb     = ws;               // 16384
  float* cnorm  = ws + 16384;       // 256
  float* sums   = ws + 16640;       // 16384
  float* counts = ws + 33024;       // 256
  float* dacc   = ws + 33280;       // 1   (sums..dacc contiguous: one zero pass)
  float* macc   = ws + 33281;       // 64
  float* sxx    = ws + 33345;       // 1
  float* state  = ws + 33346;       // 3: {prev, dist, done}

  (void)hipFuncSetAttribute(reinterpret_cast<const void*>(lbg_estep),
                            hipFuncAttributeMaxDynamicSharedMemorySize, 70656);

  // init: mean + sum of squares, codebook/state init
  zero_f<<<1, 256, 0, stream>>>(macc, 65);
  lbg_meansq<<<256, 256, 0, stream>>>(x, macc, sxx);
  lbg_initcb<<<64, 256, 0, stream>>>(cb, macc, state);

  // JAX Threefry key chain on host (data-independent => graph-capture safe)
  uint32_t base0 = 0u, base1 = 42u;   // jax.random.key(42)
  for (int split = 0; split < 8; ++split) {
    uint32_t ks0, ks1;
    tf2x32(base0, base1, 0u, (uint32_t)split, ks0, ks1);   // fold_in(base, split)
    uint32_t a0, a1, b0, b1;
    tf2x32(ks0, ks1, 0u, 2u, a0, a1);                      // split(key_s)
    tf2x32(ks0, ks1, 1u, 3u, b0, b1);
    uint32_t kr0 = a0, kr1 = b0;                           // kr    = keys[0]
    uint32_t kl0 = a1, kl1 = b1;                           // kloop = keys[1]

    const int curr_old = 1 << split;
    const int curr     = curr_old * 2;
    const int curr16   = (curr + 15) & ~15;
    lbg_split<<<1, 256, 0, stream>>>(cb, state, curr_old, kr0, kr1);

    const size_t shmem = (size_t)curr16 * 260 + 2048;
    for (int n = 0; n < 5; ++n) {
      uint32_t c0, c1, d0, d1;
      tf2x32(kl0, kl1, 0u, 2u, c0, c1);                    // key, sub = split(key)
      tf2x32(kl0, kl1, 1u, 3u, d0, d1);
      uint32_t sub0 = c1, sub1 = d1;
      kl0 = c0; kl1 = d0;

      zero_f<<<66, 256, 0, stream>>>(sums, 16641);         // sums+counts+dacc
      lbg_cnorm<<<1, 256, 0, stream>>>(cb, cnorm);
      lbg_estep<<<256, 256, shmem, stream>>>(x, cb, cnorm, sums, counts, dacc,
                                             curr, curr16);
      lbg_mstep<<<1, 256, 0, stream>>>(cb, sums, counts, dacc, state, sxx,
                                       curr, n, sub0, sub1);
    }
  }
  lbg_finalize<<<65, 256, 0, stream>>>(cb, state, (float*)d_out);
}